// ImplicitLayer_68573447848370
// MI455X (gfx1250) — compile-verified
//
#include <hip/hip_runtime.h>

typedef float v2f __attribute__((ext_vector_type(2)));
typedef float v8f __attribute__((ext_vector_type(8)));

#define ALPHA 0.2f
#define KDIM  512   // IN_FEATURES (fixed by the reference)

// ---------------------------------------------------------------------------
// Kernel A: Wa[k] = sum_j W[k,j] * a[j]   (W: [WROWS, OUT_F] row-major)
// One wave32 per row; coalesced column reads + wave shuffle reduction.
// ---------------------------------------------------------------------------
__global__ void gat_compute_Wa(const float* __restrict__ W,
                               const float* __restrict__ a,
                               float* __restrict__ Wa,
                               int out_f) {
  const int lane = threadIdx.x & 31;
  const int wave = threadIdx.x >> 5;
  const int row  = blockIdx.x * (blockDim.x >> 5) + wave;

  const float* __restrict__ wrow = W + (size_t)row * out_f;
  float s = 0.0f;
  for (int j = lane; j < out_f; j += 32) {
    s += wrow[j] * a[j];
  }
  #pragma unroll
  for (int off = 16; off > 0; off >>= 1) {
    s += __shfl_down(s, off, 32);
  }
  if (lane == 0) Wa[row] = s;
}

// ---------------------------------------------------------------------------
// Kernel B: el[i] = h[i,:] . Wa[0:K],  er[i] = h[i,:] . Wa[K:2K]
// One wave computes 16 rows of BOTH el and er with V_WMMA_F32_16X16X4_F32:
//   A (16x4)  = h tile rows m0..m0+15, K-slice [k0, k0+4)
//   B (4x16)  = col 0 -> Wa_left slice, col 1 -> Wa_right slice, cols 2..15 = 0
//   D (16x16) f32 accumulator: column 0 = el, column 1 = er
// VGPR layouts per CDNA5 ISA 7.12.2:
//   A: lane = M%16, half-wave selects K{0,1} vs K{2,3}, vgpr selects in pair
//   B: lane = N%16, same K mapping (symmetric)
//   D: vgpr v holds M = v + 8*(lane>=16), N = lane%16
// Wa is staged in LDS; B columns are built with branch-free float masks so the
// inner loop is pure {b64 load, ds loads, 2 FMA, wmma} with EXEC untouched.
// ---------------------------------------------------------------------------
__global__ void gat_compute_el_er(const float* __restrict__ h,
                                  const float* __restrict__ Wa,
                                  float* __restrict__ el,
                                  float* __restrict__ er) {
  __shared__ float sWa[2 * KDIM];   // 4 KB

  for (int t = threadIdx.x; t < 2 * KDIM; t += blockDim.x) {
    sWa[t] = Wa[t];
  }
  __syncthreads();

  const int lane  = threadIdx.x & 31;
  const int wave  = threadIdx.x >> 5;
  const int m0    = (blockIdx.x * (blockDim.x >> 5) + wave) * 16;
  const int mIdx  = lane & 15;   // A: row M;  B: col N;  D: col N
  const int halfK = lane >> 4;   // selects K pair {0,1} vs {2,3}

  const float* __restrict__ hrow = h + (size_t)(m0 + mIdx) * KDIM;

  // branch-free B-column selection masks (constant per lane)
  const float selL = (mIdx == 0) ? 1.0f : 0.0f;
  const float selR = (mIdx == 1) ? 1.0f : 0.0f;

  v8f c = {};  // fp32 accumulator, zero-init

  #pragma unroll 4
  for (int k0 = 0; k0 < KDIM; k0 += 4) {
    const int kA = k0 + 2 * halfK;

    const v2f A = *(const v2f*)(hrow + kA);       // global_load_b64

    const float wl0 = sWa[kA + 0],        wl1 = sWa[kA + 1];
    const float wr0 = sWa[KDIM + kA + 0], wr1 = sWa[KDIM + kA + 1];

    v2f B;
    B[0] = wl0 * selL + wr0 * selR;
    B[1] = wl1 * selL + wr1 * selR;

    // 8 args: (neg_a, A, neg_b, B, c_mod, C, reuse_a, reuse_b)
    c = __builtin_amdgcn_wmma_f32_16x16x4_f32(false, A, false, B,
                                              (short)0, c, false, false);
  }

  // D: vgpr v -> M = (m0 + halfK*8 + v); lane's column N = mIdx
  const int mBase = m0 + halfK * 8;
  if (mIdx == 0) {
    #pragma unroll
    for (int v = 0; v < 8; ++v) el[mBase + v] = c[v];
  } else if (mIdx == 1) {
    #pragma unroll
    for (int v = 0; v < 8; ++v) er[mBase + v] = c[v];
  }
}

// ---------------------------------------------------------------------------
// Kernel C: out[i,j] = leakyrelu(el[i] + er[j]), diagonal zeroed.
// 2D grid: blockIdx.y = row i (no integer division anywhere).
// el[i] is a uniform load; er float4 loads are coalesced and cache-resident;
// one global_store_b128 per thread -> pure store-bandwidth streamer.
// ---------------------------------------------------------------------------
__global__ void gat_broadcast_lrelu(const float* __restrict__ el,
                                    const float* __restrict__ er,
                                    float* __restrict__ out,
                                    int N) {
  const unsigned i  = blockIdx.y;
  const unsigned c4 = blockIdx.x * blockDim.x + threadIdx.x;  // float4 column
  const unsigned j4 = c4 << 2;

  const float  e = el[i];
  const float4 r = ((const float4*)er)[c4];

  float4 o;
  o.x = e + r.x; o.y = e + r.y; o.z = e + r.z; o.w = e + r.w;
  o.x = (o.x > 0.0f) ? o.x : ALPHA * o.x;
  o.y = (o.y > 0.0f) ? o.y : ALPHA * o.y;
  o.z = (o.z > 0.0f) ? o.z : ALPHA * o.z;
  o.w = (o.w > 0.0f) ? o.w : ALPHA * o.w;

  // zero the diagonal element if it falls in this float4
  o.x = (i == j4 + 0u) ? 0.0f : o.x;
  o.y = (i == j4 + 1u) ? 0.0f : o.y;
  o.z = (i == j4 + 2u) ? 0.0f : o.z;
  o.w = (i == j4 + 3u) ? 0.0f : o.w;

  ((float4*)(out + (size_t)i * N))[c4] = o;
}

// ---------------------------------------------------------------------------
extern "C" void kernel_launch(void* const* d_in, const int* in_sizes, int n_in,
                              void* d_out, int out_size, void* d_ws, size_t ws_size,
                              hipStream_t stream) {
  const float* h = (const float*)d_in[0];  // [N, IN_F]
  const float* W = (const float*)d_in[1];  // [2*IN_F, OUT_F]
  const float* a = (const float*)d_in[2];  // [OUT_F, 1]

  const int IN_F  = KDIM;
  const int OUT_F = in_sizes[2];                 // 256
  const int WROWS = in_sizes[1] / OUT_F;         // 1024 = 2*IN_F
  const int N     = in_sizes[0] / IN_F;          // 8192

  float* ws  = (float*)d_ws;
  float* Wa  = ws;               // [2*IN_F]
  float* el  = ws + WROWS;       // [N]
  float* er  = el + N;           // [N]
  float* out = (float*)d_out;    // [N, N]

  // A: 8 waves/block -> WROWS/8 blocks
  gat_compute_Wa<<<WROWS / 8, 256, 0, stream>>>(W, a, Wa, OUT_F);

  // B: 8 waves/block, 16 rows per wave -> 128 rows per block
  gat_compute_el_er<<<N / 128, 256, 0, stream>>>(h, Wa, el, er);

  // C: 2D grid -> rows in y, 256 threads x (N/1024) blocks per row in x
  dim3 gridC(N / (4 * 256), N);
  gat_broadcast_lrelu<<<gridC, 256, 0, stream>>>(el, er, out, N);
}